// AERGCN_no_MHA_77678778515477
// MI455X (gfx1250) — compile-verified
//
#include <hip/hip_runtime.h>
#include <hip/hip_bf16.h>
#include <math.h>
#include <stdint.h>

// ---- problem constants (match reference) ----
#define Bn   32
#define Ln   128
#define Sn   96
#define En   768
#define Hn   300
#define Rn   41
#define HP   304      // HIDDEN padded to multiple of 16 for div storage
#define NLBL 3
#define ADJ_LD 97     // 96 + 1 pad -> LDS bank-conflict free column access

typedef float v2f __attribute__((ext_vector_type(2)));
typedef float v8f __attribute__((ext_vector_type(8)));
typedef uint32_t u32x4 __attribute__((ext_vector_type(4)));
typedef uint32_t u32x8 __attribute__((ext_vector_type(8)));

static __device__ __forceinline__ v8f wmma_f32(v2f a, v2f b, v8f c) {
    // D = A(16x4 f32) * B(4x16 f32) + C(16x16 f32)
    return __builtin_amdgcn_wmma_f32_16x16x4_f32(false, a, false, b, (short)0, c,
                                                 false, false);
}

// ------------------------------------------------------------------
// K1: textsum[b,e] = sum_l text[b,l,e]
// ------------------------------------------------------------------
__global__ __launch_bounds__(256) void k_textsum(const float* __restrict__ text,
                                                 float* __restrict__ tsum) {
    int idx = blockIdx.x * blockDim.x + threadIdx.x;   // 0 .. B*E
    if (idx >= Bn * En) return;
    int b = idx / En, e = idx % En;
    const float* p = text + (size_t)b * Ln * En + e;
    float acc = 0.f;
    for (int l = 0; l < Ln; ++l) acc += p[(size_t)l * En];
    tsum[idx] = acc;
}

// ------------------------------------------------------------------
// K2: hc_mean[b,h] = (textsum[b,:] @ lin_W)[h] / embedding_len[b] + lin_b[h]
// (mean commutes with the affine layer)
// ------------------------------------------------------------------
__global__ __launch_bounds__(320) void k_hcmean(const float* __restrict__ tsum,
                                                const int* __restrict__ masks,
                                                const float* __restrict__ linW,
                                                const float* __restrict__ linb,
                                                float* __restrict__ hcmean) {
    int b = blockIdx.x;
    __shared__ float sLen;
    if (threadIdx.x == 0) {
        int c = 0;
        for (int l = 0; l < Ln; ++l) c += masks[b * Ln + l];
        sLen = (float)c;
    }
    __syncthreads();
    int o = threadIdx.x;
    if (o < Hn) {
        const float* ts = tsum + (size_t)b * En;
        float acc = 0.f;
        for (int e = 0; e < En; ++e) acc += ts[e] * linW[(size_t)e * Hn + o];
        hcmean[(size_t)b * Hn + o] = acc / sLen + linb[o];
    }
}

// ------------------------------------------------------------------
// K3: hg0[(b,s),h] = pos_emb[pos_tags[b,s]] @ lin_W + lin_b
// WMMA f32 GEMM: M=3072 (gathered rows), K=768, N=300
// One wave per 16x16 output tile. B edge handled via clamp+mask
// (no divergent loads -> no EXEC save/restore in the K loop).
// ------------------------------------------------------------------
__global__ __launch_bounds__(128) void k_review_gemm(const int* __restrict__ tags,
                                                     const float* __restrict__ pos_emb,
                                                     const float* __restrict__ linW,
                                                     const float* __restrict__ linb,
                                                     float* __restrict__ hg0) {
    const int NT = (Hn + 15) / 16;                       // 19
    int wave = (blockIdx.x * blockDim.x + threadIdx.x) >> 5;
    int lane = threadIdx.x & 31;
    if (wave >= (Bn * Sn / 16) * NT) return;             // wave-uniform
    int mt = wave / NT, nt = wave % NT;
    int m0 = mt * 16, n0 = nt * 16;
    int lx = lane & 15;
    int hi = lane >> 4;
    int koff = hi * 2;
    int mrow = m0 + lx;                                  // global row (b*96+s)
    int tag  = tags[mrow];
    const float* arow = pos_emb + (size_t)tag * En;
    int  ncol  = n0 + lx;
    bool nok   = ncol < Hn;
    int  nc    = nok ? ncol : 0;                         // clamped column
    float nmsk = nok ? 1.f : 0.f;                        // mask for pad columns

    v8f c = {0.f, 0.f, 0.f, 0.f, 0.f, 0.f, 0.f, 0.f};
    for (int k0 = 0; k0 < En; k0 += 4) {
        int k = k0 + koff;
        v2f a = *(const v2f*)(arow + k);
        v2f bf;
        bf.x = linW[(size_t)k * Hn + nc] * nmsk;
        bf.y = linW[(size_t)(k + 1) * Hn + nc] * nmsk;
        c = wmma_f32(a, bf, c);
    }
    if (nok) {
        float bias = linb[ncol];
        int mbase = m0 + hi * 8;
        for (int v = 0; v < 8; ++v)
            hg0[(size_t)(mbase + v) * Hn + ncol] = c[v] + bias;
    }
}

// ------------------------------------------------------------------
// K4: fused RGCN per (b,r):
//   T = hg_b @ W_r            (WMMA, 16-col panels; W panel DMA'd to
//                              LDS by the Tensor Data Mover, OOB cols
//                              auto-zero -> free edge padding)
//   intm = adj[b,r] @ T       (WMMA, K=96, adj in LDS)
//   div  = intm / denom       (denom = row sums of adj, 0 -> 1)
//   scores[b,r,s] += div[s,:] . score_W   (fused shuffle reduction)
// 6 waves / block, wave w owns output rows [16w, 16w+16).
// ------------------------------------------------------------------
__global__ __launch_bounds__(192) void k_rgcn_rel(const float* __restrict__ hg_in,
                                                  const float* __restrict__ adj,
                                                  const float* __restrict__ Wl,
                                                  const float* __restrict__ swl,
                                                  const float* __restrict__ score_b,
                                                  int layer,
                                                  float* __restrict__ divout,
                                                  float* __restrict__ scores) {
    int br = blockIdx.x;
    int b = br / Rn, r = br % Rn;

    __shared__ float sAdj[Sn * ADJ_LD];   // 37,248 B
    __shared__ float sW[Hn * 16];         // 19,200 B  (W_r column panel, TDM dest)
    __shared__ float sT[Sn * 16];         //  6,144 B
    __shared__ float sDen[Sn];
    __shared__ float sScore[Sn];

    const float* adjbr = adj + ((size_t)b * Rn + r) * Sn * Sn;
    for (int i = threadIdx.x; i < Sn * Sn; i += 192) {
        int row = i / Sn, col = i % Sn;
        sAdj[row * ADJ_LD + col] = adjbr[i];
    }
    if (threadIdx.x < Sn) sScore[threadIdx.x] = 0.f;
    __syncthreads();
    if (threadIdx.x < Sn) {
        float s = 0.f;
        for (int t = 0; t < Sn; ++t) s += sAdj[threadIdx.x * ADJ_LD + t];
        sDen[threadIdx.x] = (s == 0.f) ? 1.f : s;
    }

    int wv   = threadIdx.x >> 5;          // 0..5
    int lane = threadIdx.x & 31;
    int lx   = lane & 15;
    int hi   = lane >> 4;
    int koff = hi * 2;
    int m0   = wv * 16;

    const float* X = hg_in + (size_t)b * Sn * Hn;          // [96][300]
    float* dout = divout + ((size_t)b * Rn + r) * Sn * HP; // [96][304]

    for (int nt = 0; nt < (Hn + 15) / 16; ++nt) {
        int  n0   = nt * 16;
        int  ncol = n0 + lx;
        bool nok  = ncol < Hn;
        float swv = swl[nok ? ncol : 0] * (nok ? 1.f : 0.f);

        __syncthreads();   // everyone finished reading sW / sT of prev panel

        // --- TDM: DMA W_r[:, n0:n0+16] (300x16, stride 300) into sW.
        //     tensor_dim0 = Hn-n0 so OOB columns of the last panel read 0.
        if (wv == 0) {
            uint64_t gaddr  = (uint64_t)(uintptr_t)Wl + (uint64_t)(n0 * 4);
            uint32_t ldsoff = (uint32_t)(uintptr_t)(&sW[0]);
            uint32_t tdim0  = (uint32_t)(Hn - n0);
            u32x4 g0;
            g0[0] = 1u;                                  // count=1, user desc
            g0[1] = ldsoff;                              // lds_addr
            g0[2] = (uint32_t)gaddr;                     // global_addr[31:0]
            g0[3] = (uint32_t)(gaddr >> 32) | 0x80000000u; // addr[56:32] | type=2
            u32x8 g1;
            g1[0] = 0x20000u;                            // data_size=4B
            g1[1] = tdim0 << 16;                         // tensor_dim0[15:0]
            g1[2] = ((uint32_t)Hn) << 16;                // tensor_dim1[15:0]=300
            g1[3] = 16u << 16;                           // tile_dim0=16
            g1[4] = (uint32_t)Hn;                        // tile_dim1=300
            g1[5] = (uint32_t)Hn;                        // tensor_dim0_stride=300
            g1[6] = 0u;
            g1[7] = 0u;
            asm volatile("tensor_load_to_lds %0, %1"
                         :
                         : "s"(g0), "s"(g1)
                         : "memory");
            __builtin_amdgcn_s_wait_tensorcnt(0);
        }
        __syncthreads();   // sW panel ready for all waves

        // --- phase 1: T[m0:m0+16, panel] = X @ W_r  (B from LDS, no edge test)
        v8f acc = {0.f, 0.f, 0.f, 0.f, 0.f, 0.f, 0.f, 0.f};
        const float* xrow = X + (size_t)(m0 + lx) * Hn;
        for (int k0 = 0; k0 < Hn; k0 += 4) {
            int k = k0 + koff;
            v2f a = *(const v2f*)(xrow + k);
            v2f bf;
            bf.x = sW[k * 16 + lx];
            bf.y = sW[(k + 1) * 16 + lx];
            acc = wmma_f32(a, bf, acc);
        }
        {
            int mb = m0 + hi * 8;
            for (int v = 0; v < 8; ++v) sT[(mb + v) * 16 + lx] = acc[v];
        }
        __syncthreads();   // sT ready

        // --- phase 2: intm[m0:m0+16, panel] = adj @ T  (K = 96, all LDS) ---
        v8f acc2 = {0.f, 0.f, 0.f, 0.f, 0.f, 0.f, 0.f, 0.f};
        for (int k0 = 0; k0 < Sn; k0 += 4) {
            int k = k0 + koff;
            v2f a = *(const v2f*)(&sAdj[(m0 + lx) * ADJ_LD + k]);
            v2f bf;
            bf.x = sT[k * 16 + lx];
            bf.y = sT[(k + 1) * 16 + lx];
            acc2 = wmma_f32(a, bf, acc2);
        }

        // --- normalize, store div (pad cols are exactly 0), fused score dot ---
        int mb = m0 + hi * 8;
        for (int v = 0; v < 8; ++v) {
            int   row = mb + v;
            float d   = acc2[v] / sDen[row];
            dout[(size_t)row * HP + ncol] = d;            // unconditional (HP pad)
            float sv = d * swv;
            sv += __shfl_xor(sv, 1);
            sv += __shfl_xor(sv, 2);
            sv += __shfl_xor(sv, 4);
            sv += __shfl_xor(sv, 8);
            if (lx == 0) sScore[row] += sv;               // lanes 0 & 16, distinct rows
        }
    }

    __syncthreads();
    if (threadIdx.x < Sn)
        scores[((size_t)b * Rn + r) * Sn + threadIdx.x] =
            sScore[threadIdx.x] + score_b[layer];
}

// ------------------------------------------------------------------
// K5: softmax over relations + weighted combine + ReLU
//   hg_out[b,s,o] = relu( sum_r div[b,r,s,o] * softmax_r(scores[b,r,s]) )
// ------------------------------------------------------------------
__global__ __launch_bounds__(320) void k_combine(const float* __restrict__ divin,
                                                 const float* __restrict__ scores,
                                                 float* __restrict__ hg_out) {
    int bs = blockIdx.x;
    int b = bs / Sn, s = bs % Sn;
    __shared__ float sc[Rn];
    int t = threadIdx.x;
    if (t < Rn) sc[t] = scores[((size_t)b * Rn + t) * Sn + s];
    __syncthreads();

    float m = -1e30f;
    for (int r = 0; r < Rn; ++r) m = fmaxf(m, sc[r]);
    float sum = 0.f;
    for (int r = 0; r < Rn; ++r) sum += __expf(sc[r] - m);
    float inv = 1.f / sum;

    if (t < Hn) {
        float acc = 0.f;
        for (int r = 0; r < Rn; ++r) {
            float w = __expf(sc[r] - m) * inv;
            acc += divin[(((size_t)b * Rn + r) * Sn + s) * HP + t] * w;
        }
        hg_out[((size_t)b * Sn + s) * Hn + t] = fmaxf(acc, 0.f);
    }
}

// ------------------------------------------------------------------
// K6: hg_mean, concat with hc_mean, final dense -> out[B,3]
// ------------------------------------------------------------------
__global__ __launch_bounds__(320) void k_final(const float* __restrict__ hg,
                                               const int* __restrict__ tags,
                                               const float* __restrict__ hcmean,
                                               const float* __restrict__ dW,
                                               const float* __restrict__ db,
                                               float* __restrict__ out) {
    int b = blockIdx.x;
    __shared__ float sMean[2 * Hn];
    __shared__ float sLen;
    if (threadIdx.x == 0) {
        int c = 0;
        for (int s = 0; s < Sn; ++s) c += (tags[b * Sn + s] != 0) ? 1 : 0;
        sLen = (float)c;
    }
    __syncthreads();
    int t = threadIdx.x;
    if (t < Hn) {
        float a = 0.f;
        for (int s = 0; s < Sn; ++s) a += hg[((size_t)b * Sn + s) * Hn + t];
        sMean[t]      = a / sLen;              // hg_mean first
        sMean[Hn + t] = hcmean[(size_t)b * Hn + t];
    }
    __syncthreads();
    if (t < NLBL) {
        float a = db[t];
        for (int o = 0; o < 2 * Hn; ++o) a += sMean[o] * dW[o * NLBL + t];
        out[b * NLBL + t] = a;
    }
}

// ------------------------------------------------------------------
extern "C" void kernel_launch(void* const* d_in, const int* in_sizes, int n_in,
                              void* d_out, int out_size, void* d_ws, size_t ws_size,
                              hipStream_t stream) {
    const float* text   = (const float*)d_in[0];
    const int*   masks  = (const int*)d_in[1];
    const int*   tags   = (const int*)d_in[2];
    const float* adj    = (const float*)d_in[3];
    const float* posemb = (const float*)d_in[4];
    const float* linW   = (const float*)d_in[5];
    const float* linb   = (const float*)d_in[6];
    const float* rgcnW  = (const float*)d_in[7];
    const float* scoreW = (const float*)d_in[8];
    const float* scoreb = (const float*)d_in[9];
    const float* denseW = (const float*)d_in[10];
    const float* denseb = (const float*)d_in[11];
    float* out = (float*)d_out;

    // workspace layout (floats); total ~162 MB
    const size_t DIVSZ = (size_t)Bn * Rn * Sn * HP;   // 38,289,408
    const size_t HGSZ  = (size_t)Bn * Sn * Hn;        //    921,600
    const size_t SCSZ  = (size_t)Bn * Rn * Sn;        //    125,952
    const size_t TSSZ  = (size_t)Bn * En;             //     24,576
    float* w      = (float*)d_ws;
    float* dDiv   = w;
    float* dHg0   = dDiv + DIVSZ;
    float* dHg1   = dHg0 + HGSZ;
    float* dScore = dHg1 + HGSZ;
    float* dTsum  = dScore + SCSZ;
    float* dHcm   = dTsum + TSSZ;

    const size_t WLSTRIDE = (size_t)Rn * Hn * Hn;     // per-layer rgcn_W stride

    // hc path (mean commutes with affine layer)
    k_textsum<<<(Bn * En + 255) / 256, 256, 0, stream>>>(text, dTsum);
    k_hcmean<<<Bn, 320, 0, stream>>>(dTsum, masks, linW, linb, dHcm);

    // hg0 = pos_emb[tags] @ lin_W + b     (WMMA GEMM)
    {
        int ntiles = (Bn * Sn / 16) * ((Hn + 15) / 16);   // 192*19 = 3648 waves
        int blocks = (ntiles + 3) / 4;                    // 4 waves / block
        k_review_gemm<<<blocks, 128, 0, stream>>>(tags, posemb, linW, linb, dHg0);
    }

    // RGCN layer 0
    k_rgcn_rel<<<Bn * Rn, 192, 0, stream>>>(dHg0, adj, rgcnW, scoreW, scoreb, 0,
                                            dDiv, dScore);
    k_combine<<<Bn * Sn, 320, 0, stream>>>(dDiv, dScore, dHg1);

    // RGCN layer 1
    k_rgcn_rel<<<Bn * Rn, 192, 0, stream>>>(dHg1, adj, rgcnW + WLSTRIDE,
                                            scoreW + Hn, scoreb, 1, dDiv, dScore);
    k_combine<<<Bn * Sn, 320, 0, stream>>>(dDiv, dScore, dHg0);

    // means + concat + final dense
    k_final<<<Bn, 320, 0, stream>>>(dHg0, tags, dHcm, denseW, denseb, out);
}